// LabelBfCtcAdversaryLoss_84000970375833
// MI455X (gfx1250) — compile-verified
//
#include <hip/hip_runtime.h>
#include <hip/hip_bf16.h>

typedef __attribute__((ext_vector_type(2))) float v2f;
typedef __attribute__((ext_vector_type(8))) float v8f;

#define NEGF (-1e30f)

// Problem sizes (fixed by setup_inputs)
constexpr int Bsz = 16;
constexpr int T   = 2000;
constexpr int U   = 400;   // 25 tiles of 16
constexpr int D   = 512;   // K dim
constexpr int C   = 4096;  // vocab

// ---------------------------------------------------------------------------
// Kernel 1: gather W columns (and bias) for each (set, batch) target list.
// Wg[set*16+b][k][u] = W[k][tgt[b][u]]   (layout: sb * D*U + k*U + u)
// ---------------------------------------------------------------------------
__global__ void gather_w_kernel(const float* __restrict__ W,
                                const float* __restrict__ bias,
                                const int* __restrict__ tgt1,
                                const int* __restrict__ tgt2,
                                float* __restrict__ Wg,
                                float* __restrict__ bias_g) {
  int sb  = blockIdx.x;               // 0..31  (set*16 + b)
  int set = sb >> 4, b = sb & 15;
  const int* tgt = (set == 0 ? tgt1 : tgt2) + b * U;
  float* wg = Wg + (size_t)sb * D * U;
  for (int idx = threadIdx.x; idx < D * U; idx += blockDim.x) {
    int k = idx / U, u = idx - k * U;
    wg[idx] = W[(size_t)k * C + tgt[u]];   // coalesced write, gathered read
  }
  float* bg = bias_g + sb * U;
  for (int u = threadIdx.x; u < U; u += blockDim.x) bg[u] = bias[tgt[u]];
}

// ---------------------------------------------------------------------------
// Kernel 2: WMMA GEMM fused with online logsumexp over all C columns.
// One block = 32 rows of x (async-staged into LDS), 4 waves. Each wave
// streams a 1024-column chunk as 64 N-tiles; per N-tile it computes TWO
// 16x16 WMMA accumulators (rows 0-15 and 16-31) so each B fragment load
// feeds two V_WMMA_F32_16X16X4_F32 ops (halves W traffic). Per-lane
// running (max, sumexp) streams are merged in LDS at the end -> lse[row].
// ---------------------------------------------------------------------------
__global__ void __launch_bounds__(128)
lse_wmma_kernel(const float* __restrict__ x,   // (B*T, D) row-major
                const float* __restrict__ W,   // (D, C) row-major
                const float* __restrict__ bias,
                float* __restrict__ lse) {     // (B*T)
  __shared__ float Alds[32 * D];       // 64 KB
  __shared__ float red_m[32 * 64];     // 32 rows x (16 cols * 4 waves)
  __shared__ float red_s[32 * 64];

  const int row0 = blockIdx.x * 32;

  // Async stage of the 32xD A panel: GLOBAL_LOAD_ASYNC_TO_LDS_B128
  // (CDNA5 async copy pipe; tracked by ASYNCcnt, no dest VGPRs).
  {
    uint32_t lds_base = (uint32_t)(uintptr_t)(&Alds[0]);  // low 32b = LDS offset
    uint64_t gbase = (uint64_t)(uintptr_t)(x + (size_t)row0 * D);
    for (int i = threadIdx.x; i < (32 * D) / 4; i += 128) {
      uint32_t ldsa = lds_base + (uint32_t)i * 16u;
      uint64_t ga   = gbase + (uint64_t)i * 16u;
      asm volatile("global_load_async_to_lds_b128 %0, %1, off"
                   :: "v"(ldsa), "v"(ga) : "memory");
    }
    asm volatile("s_wait_asynccnt 0x0" ::: "memory");
  }
  __syncthreads();

  const int wave = threadIdx.x >> 5;
  const int lane = threadIdx.x & 31;
  const int half = lane >> 4;          // selects K pair / row group
  const int n    = lane & 15;          // column within tile / A row index

  float rm0[8], rs0[8], rm1[8], rs1[8];
#pragma unroll
  for (int v = 0; v < 8; ++v) {
    rm0[v] = NEGF; rs0[v] = 0.f;
    rm1[v] = NEGF; rs1[v] = 0.f;
  }

  const float* arow0 = &Alds[n * D];          // A tile 0: rows 0..15
  const float* arow1 = &Alds[(n + 16) * D];   // A tile 1: rows 16..31
  const int col_base = wave * 1024;

  for (int nt = 0; nt < 64; ++nt) {
    const int col0 = col_base + nt * 16;
    const float* wb = W + col0 + n;    // this lane's B column
    v8f acc0 = {}, acc1 = {};
#pragma unroll 4
    for (int k = 0; k < D; k += 4) {
      v2f bf;                                        // B[k+2h..+1][n] — loaded once
      bf.x = wb[(size_t)(k + 2 * half) * C];
      bf.y = wb[(size_t)(k + 2 * half + 1) * C];
      v2f a0 = *(const v2f*)&arow0[k + 2 * half];    // A[m=n][k+2h..+1]
      v2f a1 = *(const v2f*)&arow1[k + 2 * half];
      acc0 = __builtin_amdgcn_wmma_f32_16x16x4_f32(
          false, a0, false, bf, (short)0, acc0, false, false);
      acc1 = __builtin_amdgcn_wmma_f32_16x16x4_f32(
          false, a1, false, bf, (short)0, acc1, false, false);
    }
    const float bn = bias[col0 + n];
#pragma unroll
    for (int v = 0; v < 8; ++v) {      // D elem v -> row 8*half+v, col n
      float val0 = acc0[v] + bn;
      float M0   = fmaxf(rm0[v], val0);
      rs0[v] = rs0[v] * __expf(rm0[v] - M0) + __expf(val0 - M0);
      rm0[v] = M0;
      float val1 = acc1[v] + bn;
      float M1   = fmaxf(rm1[v], val1);
      rs1[v] = rs1[v] * __expf(rm1[v] - M1) + __expf(val1 - M1);
      rm1[v] = M1;
    }
  }

#pragma unroll
  for (int v = 0; v < 8; ++v) {
    int r0 = 8 * half + v;
    int r1 = 16 + r0;
    int s  = wave * 16 + n;
    red_m[r0 * 64 + s] = rm0[v];
    red_s[r0 * 64 + s] = rs0[v];
    red_m[r1 * 64 + s] = rm1[v];
    red_s[r1 * 64 + s] = rs1[v];
  }
  __syncthreads();

  if (threadIdx.x < 32) {              // merge 64 (m,s) streams per row
    int r = threadIdx.x;
    float M2 = NEGF, S = 0.f;
    for (int i = 0; i < 64; ++i) {
      float mm = red_m[r * 64 + i], ss = red_s[r * 64 + i];
      float Mn = fmaxf(M2, mm);
      S = S * __expf(M2 - Mn) + ss * __expf(mm - Mn);
      M2 = Mn;
    }
    lse[row0 + r] = M2 + __logf(S);
  }
}

// ---------------------------------------------------------------------------
// Kernel 3: lp GEMM.  lp[sb][t][u] = (x_b @ Wg_sb)[t][u] + bias_g - lse[b,t]
// One wave per 16x16 output tile, K = 512 via V_WMMA_F32_16X16X4_F32.
// ---------------------------------------------------------------------------
__global__ void __launch_bounds__(32)
lp_wmma_kernel(const float* __restrict__ x,
               const float* __restrict__ Wg,
               const float* __restrict__ bias_g,
               const float* __restrict__ lse,
               float* __restrict__ lp) {
  const int tTile = blockIdx.x;        // 0..124
  const int uTile = blockIdx.y;        // 0..24
  const int sb    = blockIdx.z;        // 0..31
  const int b     = sb & 15;
  const int lane  = threadIdx.x;
  const int half  = lane >> 4;
  const int n     = lane & 15;
  const int t0    = tTile * 16, u0 = uTile * 16;

  const float* arow = x + ((size_t)b * T + t0 + n) * D;        // A row m=n
  const float* wb   = Wg + (size_t)sb * D * U + u0 + n;        // B col u0+n

  v8f acc = {};
#pragma unroll 4
  for (int k = 0; k < D; k += 4) {
    v2f a = *(const v2f*)&arow[k + 2 * half];
    v2f bf;
    bf.x = wb[(size_t)(k + 2 * half) * U];
    bf.y = wb[(size_t)(k + 2 * half + 1) * U];
    acc = __builtin_amdgcn_wmma_f32_16x16x4_f32(
        false, a, false, bf, (short)0, acc, false, false);
  }

  const float bg = bias_g[sb * U + u0 + n];
#pragma unroll
  for (int v = 0; v < 8; ++v) {
    int t = t0 + 8 * half + v;
    float val = acc[v] + bg - lse[(size_t)b * T + t];
    lp[((size_t)sb * T + t) * U + u0 + n] = val;
  }
}

// ---------------------------------------------------------------------------
// Kernel 4: blank-free CTC alpha recursion. One block per (set, batch).
// alpha[t,u] = logaddexp(alpha[t-1,u], alpha[t-1,u-1]) + lp[t,u], frozen
// once t >= act_len.  alpha in registers; u-1 shift via LDS.
// ---------------------------------------------------------------------------
__global__ void __launch_bounds__(512)
ctc_alpha_kernel(const float* __restrict__ lp,
                 const int* __restrict__ act_lens,
                 const int* __restrict__ tlen1,
                 const int* __restrict__ tlen2,
                 float* __restrict__ loss) {
  __shared__ float sh[U + 1];
  const int sb  = blockIdx.x;
  const int set = sb >> 4, b = sb & 15;
  const int u   = threadIdx.x;          // 0..511, active for u < U
  const float* lpb = lp + (size_t)sb * T * U;
  const int alen = act_lens[b];
  const int tlen = (set == 0 ? tlen1 : tlen2)[b];

  float alpha = 0.f;
  if (u < U) alpha = lpb[u] + (u == 0 ? 0.f : NEGF);

  for (int t = 1; t < T; ++t) {
    if (u < U) sh[u + 1] = alpha;
    if (u == 0) sh[0] = NEGF;
    __syncthreads();
    if (u < U) {
      float shifted = sh[u];
      float Mx = fmaxf(alpha, shifted);
      float nw = Mx + __logf(__expf(alpha - Mx) + __expf(shifted - Mx))
               + lpb[(size_t)t * U + u];
      if (t < alen) alpha = nw;          // freeze past act_len
    }
    __syncthreads();
  }
  if (u == tlen - 1) loss[sb] = -alpha;
}

// ---------------------------------------------------------------------------
// Kernel 5: batch means + adversary combination -> 3 scalars.
// ---------------------------------------------------------------------------
__global__ void finalize_kernel(const float* __restrict__ loss,
                                float* __restrict__ out) {
  if (threadIdx.x == 0) {
    float l1 = 0.f, l2 = 0.f;
    for (int i = 0; i < 16; ++i) { l1 += loss[i]; l2 += loss[16 + i]; }
    l1 *= (1.f / 16.f);
    l2 *= (1.f / 16.f);
    out[0] = l1 - 0.5f * l2;
    out[1] = l1;
    out[2] = l2;
  }
}

// ---------------------------------------------------------------------------
extern "C" void kernel_launch(void* const* d_in, const int* in_sizes, int n_in,
                              void* d_out, int out_size, void* d_ws, size_t ws_size,
                              hipStream_t stream) {
  const float* x    = (const float*)d_in[0];  // (16,2000,512)
  const float* W    = (const float*)d_in[1];  // (512,4096)
  const float* bias = (const float*)d_in[2];  // (4096)
  const int* tgt1   = (const int*)d_in[3];    // (16,400)
  const int* tgt2   = (const int*)d_in[4];    // (16,400)
  const int* alens  = (const int*)d_in[5];    // (16)
  const int* tlen1  = (const int*)d_in[6];    // (16)
  const int* tlen2  = (const int*)d_in[7];    // (16)
  float* out = (float*)d_out;                 // 3 floats

  // workspace layout (floats)
  float* ws     = (float*)d_ws;
  float* lse    = ws;                                    // 32000
  float* Wg     = lse + Bsz * T;                         // 2*16*512*400
  float* bias_g = Wg + (size_t)2 * Bsz * D * U;          // 2*16*400
  float* lossb  = bias_g + 2 * Bsz * U;                  // 32
  float* lp     = lossb + 32;                            // 2*16*2000*400

  gather_w_kernel<<<32, 256, 0, stream>>>(W, bias, tgt1, tgt2, Wg, bias_g);

  lse_wmma_kernel<<<(Bsz * T) / 32, 128, 0, stream>>>(x, W, bias, lse);

  dim3 lgrid(T / 16, U / 16, 32);
  lp_wmma_kernel<<<lgrid, 32, 0, stream>>>(x, Wg, bias_g, lse, lp);

  ctc_alpha_kernel<<<32, 512, 0, stream>>>(lp, alens, tlen1, tlen2, lossb);

  finalize_kernel<<<1, 32, 0, stream>>>(lossb, out);
}